// RegionLossV2_49890340110378
// MI455X (gfx1250) — compile-verified
//
#include <hip/hip_runtime.h>
#include <hip/hip_bf16.h>

typedef __attribute__((ext_vector_type(2))) float v2f;
typedef __attribute__((ext_vector_type(8))) float v8f;

#define BS_    8
#define CS_    20
#define NB_    160          // BS*CS
#define NA_    3
#define NTOT_  14196        // 676 + 2704 + 10816
#define OFF1_  676
#define OFF2_  3380
#define IN_    832.0f
#define THRESH_ 0.5f
#define EPS_   1e-7f
#define NBOX_  8000         // NB_*50
#define CELLS_TOTAL_ (NB_*NA_*NTOT_)   // 6,814,080
#define PHASEB_BLOCKS_  1024
#define PHASEB_THREADS_ 256
#define NACC_  9            // obj, noobj, Lx, Ly, Lw, Lh, Lconf_obj, Lconf_noobj, Lcls

// ---------------------------------------------------------------------------
// Phase A: scatter best-anchor / ignore flags (sparse; races match JAX scatter
// nondeterminism for duplicate indices).
// ---------------------------------------------------------------------------
__global__ void build_targets_kernel(const float* __restrict__ tgt,
                                     const float* __restrict__ anchors,
                                     short* __restrict__ winner,
                                     unsigned char* __restrict__ ignore_) {
    int idx = blockIdx.x * blockDim.x + threadIdx.x;
    if (idx >= NBOX_ * 3) return;
    int t = idx / 3;
    int s = idx - t * 3;
    const float* tb = tgt + t * 5;
    if (!(tb[3] > 0.0f)) return;   // invalid box -> dropped

    int fw, soff;
    if (s == 0)      { fw = 26;  soff = 0; }
    else if (s == 1) { fw = 52;  soff = OFF1_; }
    else             { fw = 104; soff = OFF2_; }
    int fh = fw;
    int b = t / 50;

    float gx = tb[1] * (float)fw;
    float gy = tb[2] * (float)fh;
    float gw = tb[3] * IN_;
    float gh = tb[4] * IN_;
    int gi = (int)gx; gi = gi < 0 ? 0 : (gi > fw - 1 ? fw - 1 : gi);
    int gj = (int)gy; gj = gj < 0 ? 0 : (gj > fh - 1 ? fh - 1 : gj);
    int loc = gj * fw + gi;

    float ious[3];
    float best_iou = -1.0f; int best_a = 0;
    #pragma unroll
    for (int a = 0; a < 3; ++a) {
        float aw = anchors[s * 6 + 2 * a];
        float ah = anchors[s * 6 + 2 * a + 1];
        float inter = fminf(gw, aw) * fminf(gh, ah);
        float iou = inter / (gw * gh + aw * ah - inter);
        ious[a] = iou;
        if (iou > best_iou) { best_iou = iou; best_a = a; }  // first max wins
    }
    #pragma unroll
    for (int a = 0; a < 3; ++a) {
        if (ious[a] > THRESH_) {
            int flat = (b * NA_ + a) * NTOT_ + soff + loc;
            ignore_[flat] = 1;
        }
    }
    int flat_best = (b * NA_ + best_a) * NTOT_ + soff + loc;
    ignore_[flat_best] = 1;          // noobj = 0 at best cell
    winner[flat_best] = (short)t;    // obj = 1, carries box id for target recompute
}

// ---------------------------------------------------------------------------
// Phase B: streaming reduction. Common path touches only conf channel +
// winner/ignore (~48 MB total). Obj cells (<= 24k) do the full coord/cls math.
// ---------------------------------------------------------------------------
__global__ void reduce_loss_kernel(const float* __restrict__ out,
                                   const float* __restrict__ tgt,
                                   const float* __restrict__ anchors,
                                   const short* __restrict__ winner,
                                   const unsigned char* __restrict__ ignore_,
                                   float* __restrict__ partials) {
    float acc[NACC_];
    #pragma unroll
    for (int i = 0; i < NACC_; ++i) acc[i] = 0.0f;

    const unsigned total  = CELLS_TOTAL_;
    const unsigned stride = gridDim.x * blockDim.x;
    for (unsigned c = blockIdx.x * blockDim.x + threadIdx.x; c < total; c += stride) {
        unsigned b = c / (unsigned)(NA_ * NTOT_);
        unsigned r = c - b * (unsigned)(NA_ * NTOT_);
        int a = (int)(r / (unsigned)NTOT_);
        int n = (int)(r - (unsigned)a * NTOT_);
        const float* base = out + ((size_t)b * 18 + (size_t)a * 6) * NTOT_ + n;
        const float* cptr = base + (size_t)4 * NTOT_;      // conf channel

        if (c + stride < total)
            __builtin_prefetch(cptr + stride, 0, 1);       // global_prefetch_b8

        float cl = *cptr;
        float p  = 1.0f / (1.0f + __expf(-cl));
        p = fminf(fmaxf(p, EPS_), 1.0f - EPS_);

        short wt = winner[c];
        unsigned char ig = ignore_[c];

        if (wt >= 0) {  // obj cell (rare)
            acc[0] += 1.0f;
            int s, fw, soff;
            if (n < OFF1_)      { s = 0; fw = 26;  soff = 0; }
            else if (n < OFF2_) { s = 1; fw = 52;  soff = OFF1_; }
            else                { s = 2; fw = 104; soff = OFF2_; }
            (void)soff;
            const float* tb = tgt + (int)wt * 5;
            float gx = tb[1] * (float)fw;
            float gy = tb[2] * (float)fw;
            float gw = tb[3] * IN_;
            float gh = tb[4] * IN_;
            float tx = gx - floorf(gx);
            float ty = gy - floorf(gy);
            float aw = anchors[s * 6 + 2 * a];
            float ah = anchors[s * 6 + 2 * a + 1];
            float tw = __logf(fmaxf(gw / aw, EPS_));
            float th = __logf(fmaxf(gh / ah, EPS_));

            float xl = base[0];
            float yl = base[(size_t)NTOT_];
            float wl = base[(size_t)2 * NTOT_];
            float hl = base[(size_t)3 * NTOT_];
            float sx = 1.0f / (1.0f + __expf(-xl));
            float sy = 1.0f / (1.0f + __expf(-yl));
            acc[2] += (sx - tx) * (sx - tx);
            acc[3] += (sy - ty) * (sy - ty);
            acc[4] += (wl - tw) * (wl - tw);
            acc[5] += (hl - th) * (hl - th);
            acc[6] += -__logf(p);

            // class loss: logsumexp over the CS (chunk) dimension, 20 entries
            float clsl = base[(size_t)5 * NTOT_];
            int Bo = (int)(b / CS_);
            float m = -3.4e38f;
            for (int cc = 0; cc < CS_; ++cc) {
                float v = out[((size_t)(Bo * CS_ + cc) * 18 + (size_t)a * 6 + 5) * NTOT_ + n];
                m = fmaxf(m, v);
            }
            float se = 0.0f;
            for (int cc = 0; cc < CS_; ++cc) {
                float v = out[((size_t)(Bo * CS_ + cc) * 18 + (size_t)a * 6 + 5) * NTOT_ + n];
                se += __expf(v - m);
            }
            acc[8] += (m + __logf(se)) - clsl;
        } else if (ig == 0) {  // noobj cell
            acc[1] += 1.0f;
            acc[7] += -__logf(1.0f - p);
        }
    }

    // wave32 butterfly reduction
    #pragma unroll
    for (int i = 0; i < NACC_; ++i) {
        float v = acc[i];
        #pragma unroll
        for (int off = 16; off > 0; off >>= 1)
            v += __shfl_xor(v, off, 32);
        acc[i] = v;
    }
    __shared__ float smem[PHASEB_THREADS_ / 32][NACC_];
    int lane = threadIdx.x & 31;
    int wave = threadIdx.x >> 5;
    if (lane == 0) {
        #pragma unroll
        for (int i = 0; i < NACC_; ++i) smem[wave][i] = acc[i];
    }
    __syncthreads();
    if (threadIdx.x < NACC_) {
        float s = 0.0f;
        #pragma unroll
        for (int w = 0; w < PHASEB_THREADS_ / 32; ++w) s += smem[w][threadIdx.x];
        partials[(size_t)blockIdx.x * NACC_ + threadIdx.x] = s;
    }
}

// ---------------------------------------------------------------------------
// Phase C: single wave, WMMA-based deterministic reduction of 1024x9 partials.
// A = 16x4 f32 chunk of the partial array, B = ones(4x16) -> D row sums; C
// accumulates over 16 chunks. Then in-lane add of 8 C VGPRs + shfl_xor(16).
// ---------------------------------------------------------------------------
__global__ void final_reduce_kernel(const float* __restrict__ partials,
                                    float* __restrict__ outv) {
    int lane = threadIdx.x;           // 32 threads, EXEC all ones (WMMA req.)
    int m16 = lane & 15;
    int khalf = (lane < 16) ? 0 : 2;  // A layout: lanes 0-15 hold K=0,1; 16-31 hold K=2,3
    v2f ones; ones.x = 1.0f; ones.y = 1.0f;

    float res[NACC_];
    for (int ch = 0; ch < NACC_; ++ch) {
        v8f cacc = {};
        #pragma unroll
        for (int chunk = 0; chunk < PHASEB_BLOCKS_ / 64; ++chunk) {
            int e0 = chunk * 64 + 4 * m16 + khalf;
            v2f aa;
            aa.x = partials[(size_t)e0 * NACC_ + ch];
            aa.y = partials[(size_t)(e0 + 1) * NACC_ + ch];
            // v_wmma_f32_16x16x4_f32: D = A x B + C, B = ones -> row sums
            cacc = __builtin_amdgcn_wmma_f32_16x16x4_f32(
                false, aa, false, ones, (short)0, cacc, false, false);
        }
        float s = cacc[0] + cacc[1] + cacc[2] + cacc[3]
                + cacc[4] + cacc[5] + cacc[6] + cacc[7];   // M = 0..7 (this half)
        s += __shfl_xor(s, 16, 32);                         // + M = 8..15
        res[ch] = s;
    }

    if (lane == 0) {
        float n_obj   = fmaxf(res[0], 1.0f);
        float n_noobj = fmaxf(res[1], 1.0f);
        float loss = (res[2] + res[3] + res[4] + res[5]) / n_obj   // COORD
                   + res[6] / n_obj                                 // OBJ_SCALE=1
                   + 100.0f * res[7] / n_noobj                      // NOOBJ_SCALE
                   + res[8] / n_obj;                                // CLASS_SCALE=1
        outv[0] = loss;
    }
}

// ---------------------------------------------------------------------------
extern "C" void kernel_launch(void* const* d_in, const int* in_sizes, int n_in,
                              void* d_out, int out_size, void* d_ws, size_t ws_size,
                              hipStream_t stream) {
    const float* out_t   = (const float*)d_in[0];
    const float* tgt     = (const float*)d_in[1];
    const float* anchors = (const float*)d_in[2];

    char* ws = (char*)d_ws;
    size_t winBytes = (size_t)CELLS_TOTAL_ * sizeof(short);          // 13.6 MB
    short* winner = (short*)ws;
    unsigned char* ignore_ = (unsigned char*)(ws + winBytes);        // 6.8 MB
    size_t ignBytes = (size_t)CELLS_TOTAL_;
    size_t partOff = (winBytes + ignBytes + 255) & ~(size_t)255;
    float* partials = (float*)(ws + partOff);                        // 36 KB

    hipMemsetAsync(winner, 0xFF, winBytes, stream);   // winner = -1 everywhere
    hipMemsetAsync(ignore_, 0x00, ignBytes, stream);

    build_targets_kernel<<<(NBOX_ * 3 + 255) / 256, 256, 0, stream>>>(
        tgt, anchors, winner, ignore_);

    reduce_loss_kernel<<<PHASEB_BLOCKS_, PHASEB_THREADS_, 0, stream>>>(
        out_t, tgt, anchors, winner, ignore_, partials);

    final_reduce_kernel<<<1, 32, 0, stream>>>(partials, (float*)d_out);
}